// MAM_31542239822261
// MI455X (gfx1250) — compile-verified
//
#include <hip/hip_runtime.h>
#include <hip/hip_bf16.h>

typedef __attribute__((ext_vector_type(2))) float v2f;
typedef __attribute__((ext_vector_type(8))) float v8f;

#define BATCH  1024
#define SEQL   200
#define DIM    64
#define NROWS  100001            // item_emb rows (vocab+1, row 0 = pad)
#define NEGC   (-1e15f)
#define NSLICE 32
#define NTILES ((NROWS + 15) / 16)              // 6251
#define TPS    ((NTILES + NSLICE - 1) / NSLICE) // 196 tiles per slice
#define FNEG   (-3.0e38f)

// ---------------- Kernel 1: attention pooling -> Q[B][64] ----------------
__global__ __launch_bounds__(64)
void mam_pool(const int* __restrict__ log_seqs,
              const float* __restrict__ item_emb,
              const float* __restrict__ attn_key,
              const float* __restrict__ pos_emb,
              float* __restrict__ Q)
{
    __shared__ float s_seq[SEQL * 65];   // padded stride 65: conflict-free
    __shared__ float s_sim[SEQL];
    __shared__ float s_key[DIM];
    __shared__ float s_red[DIM];
    const int b = blockIdx.x;
    const int d = threadIdx.x;

    s_key[d] = attn_key[d];
    for (int l = 0; l < SEQL; ++l) {
        int id = log_seqs[b * SEQL + l];
        float e = 0.0f;
        if (id != 0) e = item_emb[id * DIM + d] * 8.0f + pos_emb[l * DIM + d];
        s_seq[l * 65 + d] = e;
        if (d == 0) s_sim[l] = (id != 0) ? 0.0f : NEGC;   // additive mask
    }
    __syncthreads();

    for (int l = d; l < SEQL; l += DIM) {
        float acc = 0.0f;
        #pragma unroll
        for (int k = 0; k < DIM; ++k) acc += s_seq[l * 65 + k] * s_key[k];
        s_sim[l] += acc;
    }
    __syncthreads();

    float lm = FNEG;
    for (int l = d; l < SEQL; l += DIM) lm = fmaxf(lm, s_sim[l]);
    s_red[d] = lm;
    __syncthreads();
    if (d == 0) { float m = s_red[0]; for (int i = 1; i < DIM; ++i) m = fmaxf(m, s_red[i]); s_red[0] = m; }
    __syncthreads();
    const float mx = s_red[0];
    __syncthreads();
    float ls = 0.0f;
    for (int l = d; l < SEQL; l += DIM) { float e = __expf(s_sim[l] - mx); s_sim[l] = e; ls += e; }
    s_red[d] = ls;
    __syncthreads();
    if (d == 0) { float s = 0.0f; for (int i = 0; i < DIM; ++i) s += s_red[i]; s_red[0] = s; }
    __syncthreads();
    const float inv = 1.0f / s_red[0];

    float q = 0.0f;
    for (int l = 0; l < SEQL; ++l) q += s_sim[l] * s_seq[l * 65 + d];
    Q[b * DIM + d] = q * inv;
}

// ------- Kernel 2: WMMA GEMM over vocab + fused online softmax partials -------
// grid: (BATCH/16 batch tiles, NSLICE vocab slices), block: 256 (8 waves)
__global__ __launch_bounds__(256)
void mam_logits(const float* __restrict__ Q,
                const float* __restrict__ item_emb,
                float* __restrict__ pmax,
                float* __restrict__ psum)
{
    __shared__ float s_max[8][16];
    __shared__ float s_sum[8][16];
    const int lane   = threadIdx.x & 31;
    const int wave_s = __builtin_amdgcn_readfirstlane(threadIdx.x >> 5); // SGPR wave id
    const int h      = lane >> 4;     // lane-half: selects K pair (A/B) / M+8 (C)
    const int n16    = lane & 15;
    const int mbase  = blockIdx.x * 16;
    const int sl     = blockIdx.y;

    // A fragments: Q tile 16x64 resident in registers for the whole vocab sweep.
    // A 16x4 layout: M = lane&15; VGPRj holds K = 4*kf + 2*(lane>>4) + j.
    v2f a[16];
    #pragma unroll
    for (int kf = 0; kf < 16; ++kf)
        a[kf] = *(const v2f*)(&Q[(mbase + n16) * DIM + 4 * kf + 2 * h]);

    // per-LANE online softmax state: each lane owns columns n = t*16 + n16
    float rmax[8], rsum[8];
    #pragma unroll
    for (int r = 0; r < 8; ++r) { rmax[r] = FNEG; rsum[r] = 0.0f; }

    const int t0 = sl * TPS;
    int tend = t0 + TPS; if (tend > NTILES) tend = NTILES;   // all-scalar bounds

    int  t   = t0 + wave_s;
    bool inb = (t * 16 + n16) < NROWS;
    v2f  cur[16];
    if (t < tend) {                                          // scalar branch
        const int row = min(t * 16 + n16, NROWS - 1);
        const float* bp = item_emb + (size_t)row * DIM + 2 * h;
        #pragma unroll
        for (int kf = 0; kf < 16; ++kf) cur[kf] = *(const v2f*)(bp + 4 * kf);
    }

    for (; t < tend; t += 8) {                               // scalar loop control
        // ---- prefetch next tile's B fragments (issued before WMMA chain) ----
        const int  tn   = (t + 8 < tend) ? (t + 8) : t;
        const bool inbn = (tn * 16 + n16) < NROWS;
        v2f nxt[16];
        {
            const int row = min(tn * 16 + n16, NROWS - 1);
            const float* bp = item_emb + (size_t)row * DIM + 2 * h;
            #pragma unroll
            for (int kf = 0; kf < 16; ++kf) nxt[kf] = *(const v2f*)(bp + 4 * kf);
        }
        // ---- 16 chained WMMAs: logits tile for columns of this tile ----
        v8f c = {};
        #pragma unroll
        for (int kf = 0; kf < 16; ++kf)
            c = __builtin_amdgcn_wmma_f32_16x16x4_f32(
                    false, a[kf], false, cur[kf], (short)0, c, false, false);

        // ---- per-lane online softmax update: ONE exp per row ----
        #pragma unroll
        for (int r = 0; r < 8; ++r) {
            const float val = c[r];
            const float dd  = val - rmax[r];
            float e = __expf(dd > 0.0f ? -dd : dd);          // exp(-|d|)
            if (!inb) e = 0.0f;                              // OOB column: no contribution
            const bool gt = inb && (dd > 0.0f);
            rsum[r] = gt ? (rsum[r] * e + 1.0f) : (rsum[r] + e);
            rmax[r] = gt ? val : rmax[r];
        }
        inb = inbn;
        #pragma unroll
        for (int kf = 0; kf < 16; ++kf) cur[kf] = nxt[kf];
    }

    // ---- merge softmax partials across the 16 lanes of each half ----
    #pragma unroll
    for (int r = 0; r < 8; ++r) {
        #pragma unroll
        for (int off = 8; off > 0; off >>= 1) {
            const float om = __shfl_xor(rmax[r], off, 16);
            const float os = __shfl_xor(rsum[r], off, 16);
            const float nm = fmaxf(rmax[r], om);
            rsum[r] = rsum[r] * __expf(rmax[r] - nm) + os * __expf(om - nm);
            rmax[r] = nm;
        }
    }

    // ---- combine the 8 waves for this (batch tile, slice) ----
    if (n16 == 0) {
        #pragma unroll
        for (int r = 0; r < 8; ++r) {                        // row m = r + 8h
            s_max[wave_s][r + 8 * h] = rmax[r];
            s_sum[wave_s][r + 8 * h] = rsum[r];
        }
    }
    __syncthreads();
    if (threadIdx.x < 16) {
        const int m = threadIdx.x;
        float gm = FNEG;
        for (int w = 0; w < 8; ++w) gm = fmaxf(gm, s_max[w][m]);
        float gs = 0.0f;
        for (int w = 0; w < 8; ++w) gs += s_sum[w][m] * __expf(s_max[w][m] - gm);
        pmax[(mbase + m) * NSLICE + sl] = gm;
        psum[(mbase + m) * NSLICE + sl] = gs;
    }
}

// -------- Kernel 3: combine slices + recompute target logit, emit output --------
__global__ __launch_bounds__(256)
void mam_final(const float* __restrict__ Q,
               const float* __restrict__ item_emb,
               const int* __restrict__ pred,
               const float* __restrict__ pmax,
               const float* __restrict__ psum,
               float* __restrict__ out)
{
    const int b = blockIdx.x * blockDim.x + threadIdx.x;
    if (b >= BATCH) return;
    float gm = FNEG;
    for (int s = 0; s < NSLICE; ++s) gm = fmaxf(gm, pmax[b * NSLICE + s]);
    float gs = 0.0f;
    for (int s = 0; s < NSLICE; ++s) gs += psum[b * NSLICE + s] * __expf(pmax[b * NSLICE + s] - gm);
    // target logit = dot(Q[b], item_emb[pred[b]])  (64 MACs, trivial)
    const float* qr = Q + b * DIM;
    const float* er = item_emb + (size_t)pred[b] * DIM;
    float t = 0.0f;
    #pragma unroll
    for (int k = 0; k < DIM; ++k) t += qr[k] * er[k];
    out[b] = __expf(t - gm) / gs;
}

extern "C" void kernel_launch(void* const* d_in, const int* in_sizes, int n_in,
                              void* d_out, int out_size, void* d_ws, size_t ws_size,
                              hipStream_t stream)
{
    (void)in_sizes; (void)n_in; (void)out_size; (void)ws_size;
    const int*   log_seqs = (const int*)  d_in[0];   // [B, L]
    const int*   pred     = (const int*)  d_in[1];   // [B, 1]
    const float* item_emb = (const float*)d_in[2];   // [V+1, D]
    const float* attn_key = (const float*)d_in[3];   // [1, D]
    const float* pos_emb  = (const float*)d_in[4];   // [L, D]
    float* out = (float*)d_out;                      // [B, 1]

    char*  ws   = (char*)d_ws;
    float* Qbuf = (float*)(ws);                       // 1024*64*4 = 262144 B
    float* pmax = (float*)(ws + 262144);              // 1024*32*4 = 131072 B
    float* psum = (float*)(ws + 262144 + 131072);     // 131072 B

    mam_pool  <<<dim3(BATCH),              dim3(64),  0, stream>>>(log_seqs, item_emb, attn_key, pos_emb, Qbuf);
    mam_logits<<<dim3(BATCH / 16, NSLICE), dim3(256), 0, stream>>>(Qbuf, item_emb, pmax, psum);
    mam_final <<<dim3(BATCH / 256),        dim3(256), 0, stream>>>(Qbuf, item_emb, pred, pmax, psum, out);
}